// Mamba2_11957188952409
// MI455X (gfx1250) — compile-verified
//
#include <hip/hip_runtime.h>
#include <hip/hip_bf16.h>
#include <math.h>

// ---------------------------------------------------------------------------
// Mamba2 block for MI455X (gfx1250, wave32).
// GEMMs: V_WMMA_F32_16X16X4_F32 (full-precision fp32 matrix op) fed by
// double-buffered GLOBAL_LOAD_ASYNC_TO_LDS_B128 staging (ASYNCcnt-tracked,
// no VGPR round trip) so the matrix pipe overlaps the global->LDS traffic.
// ---------------------------------------------------------------------------

typedef float v2f __attribute__((ext_vector_type(2)));
typedef float v8f __attribute__((ext_vector_type(8)));

#define D_MODEL 2048
#define INNER   4096
#define NHEADS  64
#define HEADDIM 64
#define BATCH   2
#define SEQLEN  4096
#define MROWS   (BATCH * SEQLEN)      // 8192

#define TM 128
#define TN 128
#define TK 64
#define LDSS 68                        // 64 + 4 pad: rows stay 16B aligned, banks spread

// Async stage of one 128x64 fp32 tile pair (A,B) into LDS: 16 B128 ops/thread.
__device__ __forceinline__ void stage_tile_async(
    float* __restrict__ as, float* __restrict__ bs,
    const float* __restrict__ X, const float* __restrict__ W,
    int m0, int n0, int k0, int K, int tid)
{
#pragma unroll
    for (int r = 0; r < 8; ++r) {
        const int idx = tid + r * 256;       // 0..2047 float4 slots
        const int row = idx >> 4;
        const int c4  = (idx & 15) << 2;
        const float* ga = X + (size_t)(m0 + row) * K + k0 + c4;
        const unsigned la = (unsigned)(uintptr_t)(as + row * LDSS + c4);
        asm volatile("global_load_async_to_lds_b128 %0, %1, off"
                     :: "v"(la), "v"(ga) : "memory");
        const float* gb = W + (size_t)(n0 + row) * K + k0 + c4;
        const unsigned lb = (unsigned)(uintptr_t)(bs + row * LDSS + c4);
        asm volatile("global_load_async_to_lds_b128 %0, %1, off"
                     :: "v"(lb), "v"(gb) : "memory");
    }
}

// O[m,n] = sum_k X[m,k] * W[n,k]   (X: MxK row-major, W: NxK row-major)
// Dims are exact multiples of the tile sizes for all three GEMMs.
__global__ __launch_bounds__(256) void gemm_xwT_wmma(
    const float* __restrict__ X, const float* __restrict__ W,
    float* __restrict__ O, int M, int N, int K)
{
    __shared__ float As[2][TM * LDSS];
    __shared__ float Bs[2][TN * LDSS];

    const int tid  = threadIdx.x;
    const int lane = tid & 31;
    const int wave = tid >> 5;
    const int l16  = lane & 15;
    const int half = lane >> 4;        // 0: K={0,1} rows/cols, 1: K={2,3}

    const int m0 = blockIdx.y * TM;
    const int n0 = blockIdx.x * TN;

    const int wm = (wave >> 2) * 64;   // 2 waves along M
    const int wn = (wave & 3)  * 32;   // 4 waves along N

    v8f acc[4][2] = {};

    const int nt = K / TK;
    stage_tile_async(As[0], Bs[0], X, W, m0, n0, 0, K, tid);

    for (int kt = 0; kt < nt; ++kt) {
        const int cur = kt & 1;
        // Wait for this wave's async stage of the current tile; barrier makes
        // every wave's stage visible to the whole workgroup.
        asm volatile("s_wait_asynccnt 0x0" ::: "memory");
        __syncthreads();

        if (kt + 1 < nt)
            stage_tile_async(As[cur ^ 1], Bs[cur ^ 1], X, W,
                             m0, n0, (kt + 1) * TK, K, tid);

        const float* __restrict__ asb = As[cur];
        const float* __restrict__ bsb = Bs[cur];
#pragma unroll 4
        for (int k4 = 0; k4 < TK; k4 += 4) {
            const int kb = k4 + 2 * half;
            v2f af[4], bf[2];
#pragma unroll
            for (int i = 0; i < 4; ++i)
                af[i] = *(const v2f*)(asb + (wm + i * 16 + l16) * LDSS + kb);
#pragma unroll
            for (int j = 0; j < 2; ++j)
                bf[j] = *(const v2f*)(bsb + (wn + j * 16 + l16) * LDSS + kb);
#pragma unroll
            for (int i = 0; i < 4; ++i)
#pragma unroll
                for (int j = 0; j < 2; ++j)
                    acc[i][j] = __builtin_amdgcn_wmma_f32_16x16x4_f32(
                        false, af[i], false, bf[j], (short)0, acc[i][j],
                        false, false);
        }
        // All reads of buffer `cur` done before it is restaged at kt+2.
        __syncthreads();
    }

    // C/D layout: vgpr e -> M = e + 8*half, lane l16 -> N.
#pragma unroll
    for (int i = 0; i < 4; ++i)
#pragma unroll
        for (int j = 0; j < 2; ++j) {
            const int n = n0 + wn + j * 16 + l16;
#pragma unroll
            for (int e = 0; e < 8; ++e) {
                const int m = m0 + wm + i * 16 + e + 8 * half;
                O[(size_t)m * N + n] = acc[i][j][e];
            }
        }
}

__device__ __forceinline__ float silu_f(float v) {
    return v / (1.f + __expf(-v));
}

// dt[b,l,h] = softplus( sum_p silu(conv(proj))[b,l,h,p] * dt_w[h,p] + dt_b[h] )
// One wave per (b,l,h); lane handles p = lane and p = lane+32.
__global__ __launch_bounds__(256) void dt_kernel(
    const float* __restrict__ proj, const float* __restrict__ conv_w,
    const float* __restrict__ conv_b, const float* __restrict__ dt_w,
    const float* __restrict__ dt_b, float* __restrict__ dt_out)
{
    const int gw   = (blockIdx.x * blockDim.x + threadIdx.x) >> 5;
    const int lane = threadIdx.x & 31;
    const int h = gw & 63;
    const int t = (gw >> 6) & (SEQLEN - 1);
    const int b = gw >> 18;            // gw / (64*4096)

    float sum = 0.f;
#pragma unroll
    for (int pp = 0; pp < 2; ++pp) {
        const int p = lane + 32 * pp;
        const int c = h * HEADDIM + p;
        float a = conv_b[c];
#pragma unroll
        for (int k = 0; k < 4; ++k) {
            const int tt = t - 3 + k;
            const float xv = (tt >= 0)
                ? proj[((size_t)b * SEQLEN + tt) * INNER + c] : 0.f;
            a += conv_w[c * 4 + k] * xv;
        }
        sum += silu_f(a) * dt_w[c];
    }
#pragma unroll
    for (int off = 16; off; off >>= 1) sum += __shfl_xor(sum, off, 32);
    if (lane == 0) {
        const float v = sum + dt_b[h];
        dt_out[((size_t)b * SEQLEN + t) * NHEADS + h] =
            (v > 20.f) ? v : log1pf(__expf(v));
    }
}

#define CHUNK  64
#define NCHUNK 64                     // SEQLEN / CHUNK
#define NCHAN  (BATCH * INNER)        // 8192 channels (b,h,p)

// Pass 1: per-(channel, chunk) aggregates (prod a, folded u).
__global__ __launch_bounds__(256) void scan_pass1(
    const float* __restrict__ proj, const float* __restrict__ conv_w,
    const float* __restrict__ conv_b, const float* __restrict__ dt,
    const float* __restrict__ A_log, const float* __restrict__ Bm,
    float* __restrict__ Aagg, float* __restrict__ Uagg)
{
    const int g   = blockIdx.x * blockDim.x + threadIdx.x;  // NCHAN*NCHUNK
    const int chn = g & (NCHAN - 1);                        // b*4096 + h*64 + p
    const int ch  = g >> 13;
    const int c   = chn & (INNER - 1);
    const int h   = c >> 6;
    const int b   = chn >> 12;
    const int t0  = ch * CHUNK;

    const float Acoef = -__expf(A_log[c]);
    const float Bc = Bm[c];
    const float w0 = conv_w[c*4+0], w1 = conv_w[c*4+1];
    const float w2 = conv_w[c*4+2], w3 = conv_w[c*4+3];
    const float bias = conv_b[c];

    const float* pb = proj + (size_t)b * SEQLEN * INNER + c;
    float x0 = (t0 - 3 >= 0) ? pb[(size_t)(t0 - 3) * INNER] : 0.f;
    float x1 = (t0 - 2 >= 0) ? pb[(size_t)(t0 - 2) * INNER] : 0.f;
    float x2 = (t0 - 1 >= 0) ? pb[(size_t)(t0 - 1) * INNER] : 0.f;

    float aprod = 1.f, uacc = 0.f;
    for (int i = 0; i < CHUNK; ++i) {
        const int t = t0 + i;
        const float x3 = pb[(size_t)t * INNER];
        const float Xv = silu_f(bias + w0*x0 + w1*x1 + w2*x2 + w3*x3);
        const float dtv = dt[((size_t)b * SEQLEN + t) * NHEADS + h];
        const float a = __expf(dtv * Acoef);
        const float u = dtv * Bc * Xv;
        aprod *= a;
        uacc = a * uacc + u;
        x0 = x1; x1 = x2; x2 = x3;
    }
    Aagg[(size_t)ch * NCHAN + chn] = aprod;
    Uagg[(size_t)ch * NCHAN + chn] = uacc;
}

// Pass 2: serial scan over the 64 chunk aggregates per channel.
__global__ __launch_bounds__(256) void scan_pass2(
    const float* __restrict__ Aagg, const float* __restrict__ Uagg,
    float* __restrict__ carry)
{
    const int chn = blockIdx.x * blockDim.x + threadIdx.x;  // NCHAN
    float s = 0.f;
    for (int ch = 0; ch < NCHUNK; ++ch) {
        carry[(size_t)ch * NCHAN + chn] = s;
        s = Aagg[(size_t)ch * NCHAN + chn] * s + Uagg[(size_t)ch * NCHAN + chn];
    }
}

// Pass 3: replay with carry, y = C*s + D*X, combined = y * sigmoid(gate).
// Writes combined in-place over the gate-linear buffer (same index).
__global__ __launch_bounds__(256) void scan_pass3(
    const float* __restrict__ proj, const float* __restrict__ conv_w,
    const float* __restrict__ conv_b, const float* __restrict__ dt,
    const float* __restrict__ A_log, const float* __restrict__ Bm,
    const float* __restrict__ Cm, const float* __restrict__ Dm,
    const float* __restrict__ carry, float* __restrict__ gate_combined)
{
    const int g   = blockIdx.x * blockDim.x + threadIdx.x;
    const int chn = g & (NCHAN - 1);
    const int ch  = g >> 13;
    const int c   = chn & (INNER - 1);
    const int h   = c >> 6;
    const int b   = chn >> 12;
    const int t0  = ch * CHUNK;

    const float Acoef = -__expf(A_log[c]);
    const float Bc = Bm[c], Cc = Cm[c], Dc = Dm[c];
    const float w0 = conv_w[c*4+0], w1 = conv_w[c*4+1];
    const float w2 = conv_w[c*4+2], w3 = conv_w[c*4+3];
    const float bias = conv_b[c];

    const float* pb = proj + (size_t)b * SEQLEN * INNER + c;
    float x0 = (t0 - 3 >= 0) ? pb[(size_t)(t0 - 3) * INNER] : 0.f;
    float x1 = (t0 - 2 >= 0) ? pb[(size_t)(t0 - 2) * INNER] : 0.f;
    float x2 = (t0 - 1 >= 0) ? pb[(size_t)(t0 - 1) * INNER] : 0.f;

    float s = carry[(size_t)ch * NCHAN + chn];
    for (int i = 0; i < CHUNK; ++i) {
        const int t = t0 + i;
        const float x3 = pb[(size_t)t * INNER];
        const float Xv = silu_f(bias + w0*x0 + w1*x1 + w2*x2 + w3*x3);
        const float dtv = dt[((size_t)b * SEQLEN + t) * NHEADS + h];
        const float a = __expf(dtv * Acoef);
        s = a * s + dtv * Bc * Xv;
        const float y = Cc * s + Dc * Xv;
        const size_t idx = ((size_t)b * SEQLEN + t) * INNER + c;
        const float gl = gate_combined[idx];
        gate_combined[idx] = y * (1.f / (1.f + __expf(-gl)));
        x0 = x1; x1 = x2; x2 = x3;
    }
}

extern "C" void kernel_launch(void* const* d_in, const int* in_sizes, int n_in,
                              void* d_out, int out_size, void* d_ws, size_t ws_size,
                              hipStream_t stream)
{
    const float* hidden  = (const float*)d_in[0];
    const float* in_w    = (const float*)d_in[1];
    const float* gate_w  = (const float*)d_in[2];
    const float* out_w   = (const float*)d_in[3];
    const float* conv_w  = (const float*)d_in[4];
    const float* conv_b  = (const float*)d_in[5];
    const float* dt_w    = (const float*)d_in[6];
    const float* dt_b    = (const float*)d_in[7];
    const float* A_log   = (const float*)d_in[8];
    const float* Bm      = (const float*)d_in[9];
    const float* Cm      = (const float*)d_in[10];
    const float* Dm      = (const float*)d_in[11];
    float* out = (float*)d_out;

    // Workspace carve-up (floats).
    float* proj  = (float*)d_ws;                       // MROWS*INNER
    float* gateC = proj  + (size_t)MROWS * INNER;      // MROWS*INNER (gate -> combined)
    float* dtbuf = gateC + (size_t)MROWS * INNER;      // MROWS*NHEADS
    float* Aagg  = dtbuf + (size_t)MROWS * NHEADS;     // NCHUNK*NCHAN
    float* Uagg  = Aagg  + (size_t)NCHUNK * NCHAN;
    float* carry = Uagg  + (size_t)NCHUNK * NCHAN;

    // 1) projected = hidden @ in_proj_w^T    (M=8192, N=4096, K=2048)
    gemm_xwT_wmma<<<dim3(INNER / TN, MROWS / TM), 256, 0, stream>>>(
        hidden, in_w, proj, MROWS, INNER, D_MODEL);
    // 2) gate linear = hidden @ gate_proj_w^T
    gemm_xwT_wmma<<<dim3(INNER / TN, MROWS / TM), 256, 0, stream>>>(
        hidden, gate_w, gateC, MROWS, INNER, D_MODEL);

    // 3) dt  (one wave per (b,l,h))
    {
        const long waves = (long)MROWS * NHEADS;           // 524288
        dt_kernel<<<(waves * 32) / 256, 256, 0, stream>>>(
            proj, conv_w, conv_b, dt_w, dt_b, dtbuf);
    }

    // 4-6) chunked scan
    scan_pass1<<<((size_t)NCHAN * NCHUNK) / 256, 256, 0, stream>>>(
        proj, conv_w, conv_b, dtbuf, A_log, Bm, Aagg, Uagg);
    scan_pass2<<<NCHAN / 256, 256, 0, stream>>>(Aagg, Uagg, carry);
    scan_pass3<<<((size_t)NCHAN * NCHUNK) / 256, 256, 0, stream>>>(
        proj, conv_w, conv_b, dtbuf, A_log, Bm, Cm, Dm, carry, gateC);

    // 7) out = combined @ out_proj_w^T   (M=8192, N=2048, K=4096)
    gemm_xwT_wmma<<<dim3(D_MODEL / TN, MROWS / TM), 256, 0, stream>>>(
        gateC, out_w, out, MROWS, D_MODEL, INNER);
}